// MNN_45896020525500
// MI455X (gfx1250) — compile-verified
//
#include <hip/hip_runtime.h>

// Hit-or-miss morphological transform for MI455X (gfx1250).
// x:(16,8,128,128) f32, K_hit/K_miss:(32,8,5,5) f32 -> out:(16,32,124,124) f32.
// Compute-bound on VALU (min/max is not WMMA-expressible); uses CDNA5 TDM
// (tensor_load_to_lds + s_wait_tensorcnt) to stage input tiles into LDS,
// scalar (SMEM) K operands, and v_min3/v_max3-friendly reduction pairing.

#define IC   8
#define OC   32
#define KS   5
#define HH   128
#define FH   124          // 128 - 5 + 1
#define TILE 16
#define TW   (TILE + KS - 1)   // 20
#define OCH  4                 // output channels per accumulation chunk

typedef unsigned int       u32;
typedef unsigned long long u64;
typedef __attribute__((ext_vector_type(4))) u32 v4u;
typedef __attribute__((ext_vector_type(8))) int v8i;
typedef __attribute__((ext_vector_type(4))) int v4i;

#if defined(__gfx1250__) && __has_builtin(__builtin_amdgcn_sched_barrier)
#define SCHED_FENCE() __builtin_amdgcn_sched_barrier(0)
#else
#define SCHED_FENCE() ((void)0)
#endif

__global__ __launch_bounds__(256) void hitmiss_kernel(
    const float* __restrict__ x,
    const float* __restrict__ Kh,
    const float* __restrict__ Km,
    float* __restrict__ out)
{
    __shared__ float tile[IC * TW * TW];   // 8 * 20 * 20 * 4B = 12.8 KB

    const int tx   = threadIdx.x & 15;
    const int ty   = threadIdx.x >> 4;
    const int col0 = blockIdx.x * TILE;
    const int row0 = blockIdx.y * TILE;
    const int b    = blockIdx.z;

    // ---------------- stage input tile into LDS via TDM ----------------
#if defined(__gfx1250__) && __has_builtin(__builtin_amdgcn_tensor_load_to_lds)
    // Scalar (wave-uniform) branch: only wave 0 issues the TDM.
    if (__builtin_amdgcn_readfirstlane(threadIdx.x) < 32) {
        u64 gaddr = (u64)x +
            ((u64)b * (u64)(IC * HH * HH) + (u64)row0 * HH + (u64)col0) * 4ull;
        u32 lds_base =
            (u32)(size_t)(__attribute__((address_space(3))) float*)tile;

        // D# group 0: count=1, lds_addr, 57-bit global_addr, type=2 ("image")
        v4u g0;
        g0.x = 1u;
        g0.y = lds_base;
        g0.z = (u32)(gaddr & 0xFFFFFFFFu);
        g0.w = (u32)((gaddr >> 32) & 0x1FFFFFFu) | (2u << 30);

        // Remaining extent from tile origin -> OOB rows/cols return zero.
        u32 dim0 = (u32)(HH - col0);
        u32 dim1 = (u32)(HH - row0);

        // D# group 1: data_size=4B (enc 2), tensor_dim0/1, tile 20x20x8,
        // dim0_stride=128, dim1_stride=128*128.
        v8i g1;
        g1[0] = (int)(2u << 16);                  // data_size = 4 bytes
        g1[1] = (int)(dim0 << 16);                // tensor_dim0[15:0]
        g1[2] = (int)(dim1 << 16);                // tensor_dim0 hi=0 | dim1 lo
        g1[3] = (int)((u32)TW << 16);             // tile_dim0 = 20
        g1[4] = (int)((u32)TW | ((u32)IC << 16)); // tile_dim1=20, tile_dim2=8
        g1[5] = (int)HH;                          // tensor_dim0_stride = 128
        g1[6] = (int)((u32)(HH * HH) << 16);      // tensor_dim1_stride lo16
        g1[7] = 0;                                // tensor_dim1_stride hi

        // D# group 2: tensor_dim2=8, tensor_dim3=1, dim2_stride, tile_dim3=0
        v4i g2;
        g2[0] = IC;
        g2[1] = 1;
        g2[2] = IC * HH * HH;
        g2[3] = 0;

        v4i g3 = {0, 0, 0, 0};

#if __has_include(<hip/amd_detail/amd_gfx1250_TDM.h>)
        v8i gz = {0, 0, 0, 0, 0, 0, 0, 0};
        __builtin_amdgcn_tensor_load_to_lds(g0, g1, g2, g3, gz, 0);
#else
        __builtin_amdgcn_tensor_load_to_lds(g0, g1, g2, g3, 0);
#endif
#if __has_builtin(__builtin_amdgcn_s_wait_tensorcnt)
        __builtin_amdgcn_s_wait_tensorcnt(0);
#endif
    }
#else
    // Fallback: cooperative vector loads (also used for the host pass).
    for (int idx = threadIdx.x; idx < IC * TW * TW; idx += 256) {
        int c   = idx / (TW * TW);
        int rem = idx % (TW * TW);
        int r   = rem / TW;
        int cc  = rem % TW;
        int gr  = row0 + r;  if (gr > HH - 1) gr = HH - 1;
        int gc  = col0 + cc; if (gc > HH - 1) gc = HH - 1;
        tile[idx] = x[((size_t)b * IC + c) * (HH * HH) + (size_t)gr * HH + gc];
    }
#endif
    __syncthreads();

    const int  gi    = row0 + ty;
    const int  gj    = col0 + tx;
    const bool valid = (gi < FH) && (gj < FH);

    // ---------------- compute ----------------
    // 8 chunks of 4 output channels. Per (chunk, in-channel): window in VGPRs;
    // K values are block-uniform -> scalar s_load operands of v_sub_f32.
    // Hit and miss passes are separated by sched barriers so at most ~100
    // K dwords are live at once (no SGPR->VGPR spilling).
#pragma unroll 1
    for (int oc = 0; oc < OC / OCH; ++oc) {
        float hm[OCH], mm[OCH];
#pragma unroll
        for (int i = 0; i < OCH; ++i) {
            hm[i] =  __builtin_inff();
            mm[i] = -__builtin_inff();
        }

#pragma unroll 1
        for (int c = 0; c < IC; ++c) {
            float win[KS * KS];
#pragma unroll
            for (int u = 0; u < KS; ++u)
#pragma unroll
                for (int v = 0; v < KS; ++v)
                    win[u * KS + v] =
                        tile[c * (TW * TW) + (ty + u) * TW + (tx + v)];

            // ---- hit pass (only K_hit chunk live) ----
#pragma unroll
            for (int i = 0; i < OCH; ++i) {
                const int o = oc * OCH + i;
                const float* __restrict__ kh = Kh + (o * IC + c) * (KS * KS);
                float h = hm[i];
#pragma unroll
                for (int t = 0; t < 24; t += 2) {
                    h = __builtin_fminf(
                        h, __builtin_fminf(win[t] - kh[t], win[t + 1] - kh[t + 1]));
                }
                h = __builtin_fminf(h, win[24] - kh[24]);
                hm[i] = h;
            }
            SCHED_FENCE();

            // ---- miss pass (only K_miss chunk live) ----
#pragma unroll
            for (int i = 0; i < OCH; ++i) {
                const int o = oc * OCH + i;
                const float* __restrict__ km = Km + (o * IC + c) * (KS * KS);
                float m = mm[i];
#pragma unroll
                for (int t = 0; t < 24; t += 2) {
                    m = __builtin_fmaxf(
                        m, __builtin_fmaxf(win[t] - km[t], win[t + 1] - km[t + 1]));
                }
                m = __builtin_fmaxf(m, win[24] - km[24]);
                mm[i] = m;
            }
            SCHED_FENCE();
        }

        if (valid) {
#pragma unroll
            for (int i = 0; i < OCH; ++i) {
                const int o = oc * OCH + i;
                out[(((size_t)b * OC + o) * FH + gi) * FH + gj] = hm[i] - mm[i];
            }
        }
    }
}

extern "C" void kernel_launch(void* const* d_in, const int* in_sizes, int n_in,
                              void* d_out, int out_size, void* d_ws, size_t ws_size,
                              hipStream_t stream)
{
    (void)in_sizes; (void)n_in; (void)out_size; (void)d_ws; (void)ws_size;
    const float* x  = (const float*)d_in[0];
    const float* Kh = (const float*)d_in[1];
    const float* Km = (const float*)d_in[2];
    float*       o  = (float*)d_out;

    dim3 grid((FH + TILE - 1) / TILE, (FH + TILE - 1) / TILE, 16);  // 8,8,16
    dim3 block(256);
    hipLaunchKernelGGL(hitmiss_kernel, grid, block, 0, stream, x, Kh, Km, o);
}